// CycleBalanceLoss_25546465477019
// MI455X (gfx1250) — compile-verified
//
#include <hip/hip_runtime.h>
#include <hip/hip_bf16.h>

typedef float v2f __attribute__((ext_vector_type(2)));
typedef float v8f __attribute__((ext_vector_type(8)));

#define ALPHA_F 0.7f
#define B_SZ 64
#define L_SZ 128
#define N_SZ 1024

// ---------------------------------------------------------------------------
// Kernel A: one wave per 16 rows (rows = (b,l) pairs). Pass 1: row max,
// argmax (first-max tie-break), target logit. Pass 2: sum of exp(x - max)
// accumulated on the matrix pipe via V_WMMA_F32_16X16X4_F32 with B == ones
// (layout-independent in B; A uses the documented 16x4 f32 striping:
// lanes 0-15 hold K={0,1} in v[0],v[1]; lanes 16-31 hold K={2,3}).
// ---------------------------------------------------------------------------
__global__ __launch_bounds__(32) void row_softmax_argmax_kernel(
    const float* __restrict__ logits,   // [8192, 1024]
    const int*   __restrict__ targets,  // [8192]
    float*       __restrict__ nll,      // [8192]
    int*         __restrict__ amax)     // [8192]
{
    const int lane = threadIdx.x;          // 0..31
    const int m    = lane & 15;            // row within the 16-row tile
    const int half = lane >> 4;            // 0 -> K%4 in {0,1}, 1 -> {2,3}
    const int row  = blockIdx.x * 16 + m;  // 0..8191
    const float* base = logits + (size_t)row * N_SZ + 2 * half;
    const int tgt = targets[row];

    // ---- pass 1: max / argmax / target logit over this lane's K-subset ----
    float bmax = -__builtin_inff();
    int   bidx = 0;
    float tlog = 0.0f;
    #pragma unroll 4
    for (int kk = 0; kk < N_SZ / 4; ++kk) {
        const float2 x = *(const float2*)(base + 4 * kk);
        const int k0 = 4 * kk + 2 * half;
        if (x.x > bmax) { bmax = x.x; bidx = k0; }       // strict > keeps first max
        if (x.y > bmax) { bmax = x.y; bidx = k0 + 1; }
        if (k0     == tgt) tlog = x.x;                   // exactly one lane owns tgt
        if (k0 + 1 == tgt) tlog = x.y;
    }
    // combine the two K-halves of each row (lanes l <-> l+16)
    const float omax = __shfl_xor(bmax, 16, 32);
    const int   oidx = __shfl_xor(bidx, 16, 32);
    tlog += __shfl_xor(tlog, 16, 32);                    // non-owner contributed 0
    if (omax > bmax || (omax == bmax && oidx < bidx)) { bmax = omax; bidx = oidx; }

    // ---- pass 2: WMMA-accumulated sum of exp(x - rowmax) ----
    v8f c = {0.f, 0.f, 0.f, 0.f, 0.f, 0.f, 0.f, 0.f};
    v2f ones; ones[0] = 1.0f; ones[1] = 1.0f;            // B all-ones
    #pragma unroll 2
    for (int kk = 0; kk < N_SZ / 4; ++kk) {
        const float2 x = *(const float2*)(base + 4 * kk);
        v2f a;
        a[0] = __expf(x.x - bmax);
        a[1] = __expf(x.y - bmax);
        // D[m][n] = sum_k A[m][k] + C[m][n]  (uniform control flow, EXEC all-1s)
        c = __builtin_amdgcn_wmma_f32_16x16x4_f32(false, a, false, ones,
                                                  (short)0, c, false, false);
    }

    // C/D layout: vgpr i, lanes 0-15 -> M=i; lanes 16-31 -> M=8+i.  Column N=0
    // of each half-wave carries the per-row sums; spill through LDS.
    __shared__ float lsum[16];
    if (lane == 0) {
        #pragma unroll
        for (int i = 0; i < 8; ++i) lsum[i] = c[i];
    }
    if (lane == 16) {
        #pragma unroll
        for (int i = 0; i < 8; ++i) lsum[8 + i] = c[i];
    }
    __syncthreads();

    if (half == 0) {
        const float sumexp = lsum[m];
        nll[row]  = __logf(sumexp) + bmax - tlog;   // -log_softmax[target]
        amax[row] = bidx;
    }
}

// ---------------------------------------------------------------------------
// Kernel B: per-batch gather of adj[b, idx[l], idx[l+1]], deterministic tree
// reduce, abs.
// ---------------------------------------------------------------------------
__global__ __launch_bounds__(L_SZ) void balance_kernel(
    const float* __restrict__ adj,    // [64, 1024, 1024]
    const int*   __restrict__ amax,   // [8192]
    float*       __restrict__ bal)    // [64]
{
    const int b = blockIdx.x;
    const int l = threadIdx.x;
    __shared__ float red[L_SZ];
    float w = 0.0f;
    if (l < L_SZ - 1) {
        const int s = amax[b * L_SZ + l];
        const int d = amax[b * L_SZ + l + 1];
        w = adj[((size_t)b << 20) + ((size_t)s << 10) + (size_t)d];
    }
    red[l] = w;
    __syncthreads();
    #pragma unroll
    for (int off = L_SZ / 2; off >= 1; off >>= 1) {
        if (l < off) red[l] += red[l + off];
        __syncthreads();
    }
    if (l == 0) bal[b] = fabsf(red[0]);
}

// ---------------------------------------------------------------------------
// Kernel C: deterministic final reduction + blend.
// ---------------------------------------------------------------------------
__global__ __launch_bounds__(256) void final_kernel(
    const float* __restrict__ nll,   // [8192]
    const float* __restrict__ bal,   // [64]
    float*       __restrict__ out)   // [1]
{
    __shared__ float red[256];
    const int t = threadIdx.x;

    float s = 0.0f;
    for (int i = t; i < B_SZ * L_SZ; i += 256) s += nll[i];  // fixed order
    red[t] = s;
    __syncthreads();
    #pragma unroll
    for (int off = 128; off >= 1; off >>= 1) {
        if (t < off) red[t] += red[t + off];
        __syncthreads();
    }
    const float nll_sum = red[0];
    __syncthreads();

    red[t] = (t < B_SZ) ? bal[t] : 0.0f;
    __syncthreads();
    #pragma unroll
    for (int off = 128; off >= 1; off >>= 1) {
        if (t < off) red[t] += red[t + off];
        __syncthreads();
    }
    if (t == 0) {
        const float balance_loss = red[0] / (float)B_SZ;
        const float seq_loss     = nll_sum / (float)(B_SZ * L_SZ);
        out[0] = ALPHA_F * balance_loss + (1.0f - ALPHA_F) * seq_loss;
    }
}

// ---------------------------------------------------------------------------
extern "C" void kernel_launch(void* const* d_in, const int* in_sizes, int n_in,
                              void* d_out, int out_size, void* d_ws, size_t ws_size,
                              hipStream_t stream) {
    (void)in_sizes; (void)n_in; (void)out_size; (void)ws_size;
    const float* logits  = (const float*)d_in[0];   // [64,128,1024] f32
    const int*   targets = (const int*)  d_in[1];   // [64,128] i32 (JAX x64 off)
    const float* adj     = (const float*)d_in[2];   // [64,1024,1024] f32
    float* out = (float*)d_out;

    float* nll  = (float*)d_ws;                                   // 8192 f32
    int*   amax = (int*)((char*)d_ws + B_SZ * L_SZ * sizeof(float));   // 8192 i32
    float* bal  = (float*)((char*)d_ws + 2 * B_SZ * L_SZ * sizeof(float)); // 64 f32

    row_softmax_argmax_kernel<<<(B_SZ * L_SZ) / 16, 32, 0, stream>>>(
        logits, targets, nll, amax);
    balance_kernel<<<B_SZ, L_SZ, 0, stream>>>(adj, amax, bal);
    final_kernel<<<1, 256, 0, stream>>>(nll, bal, out);
}